// GNN31_27410481283400
// MI455X (gfx1250) — compile-verified
//
#include <hip/hip_runtime.h>
#include <hip/hip_bf16.h>
#include <math.h>

typedef __attribute__((ext_vector_type(2))) float v2f;
typedef __attribute__((ext_vector_type(8))) float v8f;

#define GAT_HEADS 6
#define GAT_LEAKY 0.2f

static __device__ __forceinline__ void atomicMaxF(float* addr, float val) {
  // IEEE-754 ordered atomic max: signed-int max for >=0, unsigned min for <0.
  if (val >= 0.0f) atomicMax((int*)addr, __float_as_int(val));
  else             atomicMin((unsigned int*)addr, __float_as_uint(val));
}

// ---------------------------------------------------------------------------
// H[n, HEADS*f_out] = X[n, f_in] @ W[HEADS, f_in, f_out]   (fp32 WMMA 16x16x4)
// One wave computes a 64x16 tile: 4 row sub-tiles share one B fragment, so
// each K-chunk is 2 B loads + 8 contiguous A loads feeding 4 WMMAs.
// Main K loop is unguarded; the f_in=11 tail clamps addresses + cndmask-zeros.
// ---------------------------------------------------------------------------
__global__ void gat_xform_wmma(const float* __restrict__ X,
                               const float* __restrict__ W,
                               float* __restrict__ H,
                               int n, int f_in, int f_out) {
  const int total     = GAT_HEADS * f_out;
  const int col_tiles = total >> 4;
  const int row_tiles = (n + 63) >> 6;                 // 64 rows per wave
  const int wave = blockIdx.x * (blockDim.x >> 5) + (threadIdx.x >> 5);
  if (wave >= row_tiles * col_tiles) return;           // uniform per wave: EXEC all-1s

  const int lane = threadIdx.x & 31;
  const int hi   = lane >> 4;        // 0: lanes 0-15, 1: lanes 16-31
  const int hi2  = hi * 2;
  const int lid  = lane & 15;

  const int ct = wave % col_tiles;
  const int rt = wave / col_tiles;
  const int row_base = rt << 6;
  const int col_base = ct << 4;
  const int head = col_base / f_out;
  const int fo   = col_base % f_out;

  const float* Wt = W + (size_t)head * f_in * f_out + fo;        // B[k][c] = Wt[k*f_out+c]
  const float* Wl = Wt + (size_t)hi2 * f_out + lid;              // per-lane B base

  // per-row A bases, address-clamped so loads are always in-bounds
  const float* Xb[4];
  #pragma unroll
  for (int t = 0; t < 4; ++t) {
    int r = row_base + t * 16 + lid;
    if (r > n - 1) r = n - 1;
    Xb[t] = X + (size_t)r * f_in;
  }

  const v8f z = {};
  v8f acc[4] = {z, z, z, z};

  const int kmain = f_in & ~3;
  int k0 = 0;
  for (; k0 < kmain; k0 += 4) {
    v2f b;
    b.x = Wl[(size_t)k0 * f_out];
    b.y = Wl[(size_t)(k0 + 1) * f_out];
    const int ka = k0 + hi2;
    #pragma unroll
    for (int t = 0; t < 4; ++t) {
      v2f a;
      a.x = Xb[t][ka];
      a.y = Xb[t][ka + 1];
      acc[t] = __builtin_amdgcn_wmma_f32_16x16x4_f32(false, a, false, b,
                                                     (short)0, acc[t], false, false);
    }
  }
  if (k0 < f_in) {                                     // K tail (layer 1: 3 leftover)
    const int ka  = k0 + hi2;
    const int kb  = ka + 1;
    const int kac = (ka < f_in) ? ka : (f_in - 1);     // clamp address, mask value
    const int kbc = (kb < f_in) ? kb : (f_in - 1);
    v2f b;
    b.x = (ka < f_in) ? Wt[(size_t)kac * f_out + lid] : 0.0f;
    b.y = (kb < f_in) ? Wt[(size_t)kbc * f_out + lid] : 0.0f;
    float bx = Wt[(size_t)kac * f_out + lid];
    float by = Wt[(size_t)kbc * f_out + lid];
    b.x = (ka < f_in) ? bx : 0.0f;
    b.y = (kb < f_in) ? by : 0.0f;
    #pragma unroll
    for (int t = 0; t < 4; ++t) {
      float ax = Xb[t][kac];
      float ay = Xb[t][kbc];
      v2f a;
      a.x = (ka < f_in) ? ax : 0.0f;
      a.y = (kb < f_in) ? ay : 0.0f;
      acc[t] = __builtin_amdgcn_wmma_f32_16x16x4_f32(false, a, false, b,
                                                     (short)0, acc[t], false, false);
    }
  }

  // C/D layout: VGPR r -> lanes 0-15: (M=r, N=lid), lanes 16-31: (M=r+8, N=lid)
  #pragma unroll
  for (int t = 0; t < 4; ++t) {
    #pragma unroll
    for (int r = 0; r < 8; ++r) {
      const int row = row_base + t * 16 + r + hi * 8;
      if (row < n) H[(size_t)row * total + col_base + lid] = acc[t][r];
    }
  }
}

// ---------------------------------------------------------------------------
// es/ed[n, h] = dot(H[n, h, :], a_src/a_dst[h, :])
// ---------------------------------------------------------------------------
__global__ void gat_attn_terms(const float* __restrict__ H,
                               const float* __restrict__ a_src,
                               const float* __restrict__ a_dst,
                               float* __restrict__ es, float* __restrict__ ed,
                               int n, int f_out) {
  const int idx = blockIdx.x * blockDim.x + threadIdx.x;
  if (idx >= n * GAT_HEADS) return;
  const int node = idx / GAT_HEADS;
  const int h    = idx % GAT_HEADS;
  const float* hp = H + (size_t)node * GAT_HEADS * f_out + (size_t)h * f_out;
  const float* as = a_src + (size_t)h * f_out;
  const float* ad = a_dst + (size_t)h * f_out;
  float s = 0.0f, d = 0.0f;
  for (int i = 0; i < f_out; ++i) { const float v = hp[i]; s += v * as[i]; d += v * ad[i]; }
  es[idx] = s;
  ed[idx] = d;
}

__global__ void gat_fill(float* __restrict__ p, float v, long nelem) {
  const long i = (long)blockIdx.x * blockDim.x + threadIdx.x;
  if (i < nelem) p[i] = v;
}

// Pass 1: per-edge leaky-relu logit + atomic segment max over dst
__global__ void gat_edge_logits_max(const int* __restrict__ src, const int* __restrict__ dst,
                                    const float* __restrict__ es, const float* __restrict__ ed,
                                    float* __restrict__ edge_e, float* __restrict__ nodemax,
                                    int E) {
  const int idx = blockIdx.x * blockDim.x + threadIdx.x;
  if (idx >= E * GAT_HEADS) return;
  const int e = idx / GAT_HEADS;
  const int h = idx % GAT_HEADS;
  const int s = src[e], d = dst[e];
  float v = es[s * GAT_HEADS + h] + ed[d * GAT_HEADS + h];
  v = (v >= 0.0f) ? v : GAT_LEAKY * v;
  edge_e[idx] = v;
  atomicMaxF(&nodemax[d * GAT_HEADS + h], v);
}

// Pass 2: ex = exp(e - max[dst]); atomic segment sum of denominators
__global__ void gat_edge_exp_sum(const int* __restrict__ dst,
                                 float* __restrict__ edge_e,
                                 const float* __restrict__ nodemax,
                                 float* __restrict__ denom, int E) {
  const int idx = blockIdx.x * blockDim.x + threadIdx.x;
  if (idx >= E * GAT_HEADS) return;
  const int e = idx / GAT_HEADS;
  const int h = idx % GAT_HEADS;
  const int d = dst[e];
  float m = nodemax[d * GAT_HEADS + h];
  if (!(m > -3.0e38f)) m = 0.0f;           // mirror reference's isfinite() guard
  const float ex = __expf(edge_e[idx] - m);
  edge_e[idx] = ex;
  atomicAdd(&denom[d * GAT_HEADS + h], ex);
}

// Pass 3: out[dst] += alpha * h[src]   (thread = (edge, head), float4 gathers)
__global__ void gat_edge_aggregate(const int* __restrict__ src, const int* __restrict__ dst,
                                   const float* __restrict__ edge_e,
                                   const float* __restrict__ denom,
                                   const float* __restrict__ H,
                                   float* __restrict__ out, int E, int f_out) {
  const int idx = blockIdx.x * blockDim.x + threadIdx.x;
  if (idx >= E * GAT_HEADS) return;
  const int e = idx / GAT_HEADS;
  const int h = idx % GAT_HEADS;
  const int s = src[e], d = dst[e];
  const float alpha = edge_e[idx] / (denom[d * GAT_HEADS + h] + 1e-16f);
  // offsets are multiples of 16 floats -> 16B-aligned float4 gathers
  const float4* hs = (const float4*)(H   + (size_t)s * GAT_HEADS * f_out + (size_t)h * f_out);
  float*        op =                 out + (size_t)d * GAT_HEADS * f_out + (size_t)h * f_out;
  const int q = f_out >> 2;
  for (int i = 0; i < q; ++i) {
    const float4 v = hs[i];
    atomicAdd(&op[4 * i + 0], alpha * v.x);
    atomicAdd(&op[4 * i + 1], alpha * v.y);
    atomicAdd(&op[4 * i + 2], alpha * v.z);
    atomicAdd(&op[4 * i + 3], alpha * v.w);
  }
}

__global__ void gat_elu_inplace(float* __restrict__ x, long nelem) {
  const long i = (long)blockIdx.x * blockDim.x + threadIdx.x;
  if (i >= nelem) return;
  const float v = x[i];
  x[i] = (v > 0.0f) ? v : (expf(v) - 1.0f);
}

// Sum pool: g[col] = sum_n h[n, col].  blockDim.x == cols; grid-stride over nodes.
__global__ void gat_sum_pool(const float* __restrict__ h, float* __restrict__ g,
                             int n, int cols) {
  const int col = threadIdx.x;
  if (col >= cols) return;
  float acc = 0.0f;
  for (int node = blockIdx.x; node < n; node += gridDim.x)
    acc += h[(size_t)node * cols + col];
  atomicAdd(&g[col], acc);
}

// normalize(g) @ Wd + bd  -> single scalar
__global__ void gat_final_dense(const float* __restrict__ g, const float* __restrict__ Wd,
                                const float* __restrict__ bd, float* __restrict__ out,
                                int cols) {
  __shared__ float red[512];
  const int t = threadIdx.x;
  const float gv = (t < cols) ? g[t] : 0.0f;
  red[t] = gv * gv;
  __syncthreads();
  for (int s = 256; s > 0; s >>= 1) { if (t < s) red[t] += red[t + s]; __syncthreads(); }
  __shared__ float norm;
  if (t == 0) norm = fmaxf(sqrtf(red[0]), 1e-12f);
  __syncthreads();
  red[t] = (t < cols) ? (gv / norm) * Wd[t] : 0.0f;
  __syncthreads();
  for (int s = 256; s > 0; s >>= 1) { if (t < s) red[t] += red[t + s]; __syncthreads(); }
  if (t == 0) out[0] = red[0] + bd[0];
}

// ---------------------------------------------------------------------------
static inline int cdiv(long a, int b) { return (int)((a + b - 1) / b); }

extern "C" void kernel_launch(void* const* d_in, const int* in_sizes, int n_in,
                              void* d_out, int out_size, void* d_ws, size_t ws_size,
                              hipStream_t stream) {
  const float* x   = (const float*)d_in[0];
  const int*   ei  = (const int*)  d_in[1];
  const float* Ws[3]  = {(const float*)d_in[2], (const float*)d_in[5], (const float*)d_in[8]};
  const float* asA[3] = {(const float*)d_in[3], (const float*)d_in[6], (const float*)d_in[9]};
  const float* adA[3] = {(const float*)d_in[4], (const float*)d_in[7], (const float*)d_in[10]};
  const float* Wd = (const float*)d_in[11];
  const float* bd = (const float*)d_in[12];

  const int N = in_sizes[0] / 11;
  const int E = in_sizes[1] / 2;
  const int* src = ei;
  const int* dst = ei + E;

  const int fin[3]  = {11, 96, 192};
  const int fout[3] = {16, 32, 64};

  // ---- workspace carve (all fp32) -----------------------------------------
  char* w = (char*)d_ws;
  float* bufA    = (float*)w; w += (size_t)N * 384 * sizeof(float);         // xformed h
  float* bufB    = (float*)w; w += (size_t)N * 384 * sizeof(float);         // aggregated out
  float* edge_e  = (float*)w; w += (size_t)E * GAT_HEADS * sizeof(float);
  float* es      = (float*)w; w += (size_t)N * GAT_HEADS * sizeof(float);
  float* ed      = (float*)w; w += (size_t)N * GAT_HEADS * sizeof(float);
  float* nodemax = (float*)w; w += (size_t)N * GAT_HEADS * sizeof(float);
  float* denom   = (float*)w; w += (size_t)N * GAT_HEADS * sizeof(float);
  float* g       = (float*)w; w += 384 * sizeof(float);

  const float NEG_INF = -__builtin_huge_valf();

  const float* layer_in = x;
  for (int l = 0; l < 3; ++l) {
    const int fi = fin[l], fo = fout[l];
    const int total = GAT_HEADS * fo;

    // 1) dense transform via fp32 WMMA (8 waves / block, 64x16 tile per wave)
    {
      const int tiles = ((N + 63) / 64) * (total / 16);
      gat_xform_wmma<<<cdiv(tiles, 8), 256, 0, stream>>>(layer_in, Ws[l], bufA, N, fi, fo);
    }
    // 2) per-node attention terms
    gat_attn_terms<<<cdiv((long)N * GAT_HEADS, 256), 256, 0, stream>>>(
        bufA, asA[l], adA[l], es, ed, N, fo);
    // 3) init accumulators (layer input bufB already consumed by the GEMM)
    gat_fill<<<cdiv((long)N * GAT_HEADS, 256), 256, 0, stream>>>(nodemax, NEG_INF, (long)N * GAT_HEADS);
    gat_fill<<<cdiv((long)N * GAT_HEADS, 256), 256, 0, stream>>>(denom, 0.0f, (long)N * GAT_HEADS);
    gat_fill<<<cdiv((long)N * total, 256), 256, 0, stream>>>(bufB, 0.0f, (long)N * total);
    // 4) segment softmax + aggregation
    gat_edge_logits_max<<<cdiv((long)E * GAT_HEADS, 256), 256, 0, stream>>>(
        src, dst, es, ed, edge_e, nodemax, E);
    gat_edge_exp_sum<<<cdiv((long)E * GAT_HEADS, 256), 256, 0, stream>>>(
        dst, edge_e, nodemax, denom, E);
    gat_edge_aggregate<<<cdiv((long)E * GAT_HEADS, 256), 256, 0, stream>>>(
        src, dst, edge_e, denom, bufA, bufB, E, fo);
    // 5) ELU
    gat_elu_inplace<<<cdiv((long)N * total, 256), 256, 0, stream>>>(bufB, (long)N * total);

    layer_in = bufB;
  }

  // ---- sum pool -> normalize -> dense(384,1) ------------------------------
  gat_fill<<<2, 256, 0, stream>>>(g, 0.0f, 384);
  gat_sum_pool<<<512, 384, 0, stream>>>(bufB, g, N, 384);
  gat_final_dense<<<1, 512, 0, stream>>>(g, Wd, bd, (float*)d_out, 384);
}